// FeatureAct_45397804319425
// MI455X (gfx1250) — compile-verified
//
#include <hip/hip_runtime.h>
#include <hip/hip_bf16.h>
#include <math.h>

typedef __attribute__((ext_vector_type(2))) float v2f;
typedef __attribute__((ext_vector_type(8))) float v8f;

#define BATCH 2
#define C256 256
#define HW_SRC 3136   // 56*56
#define HW_POOL 784   // 28*28

// ---------------------------------------------------------------------------
// Kernel 1: fp32 WMMA GEMM   Y[b,o,n] = sum_c W[o,c] * X[b,c,n] + bias[o]
// One wave per 16x16 output tile; K=256 consumed 4 at a time with
// V_WMMA_F32_16X16X4_F32 (full fp32 precision, matching the reference dtype).
// M=256 and HW are multiples of 16 -> no partial tiles, EXEC all-1s at WMMA.
// ---------------------------------------------------------------------------
__global__ void __launch_bounds__(256)
gemm_wmma_f32(const float* __restrict__ X, const float* __restrict__ W,
              const float* __restrict__ bias, float* __restrict__ Y,
              int HW, int ntn /* HW/16 */) {
    const int wave = (blockIdx.x * blockDim.x + threadIdx.x) >> 5;
    const int lane = threadIdx.x & 31;
    const int tilesPerBatch = (C256 / 16) * ntn;
    const int b  = wave / tilesPerBatch;
    if (b >= BATCH) return;                 // wave-uniform guard
    const int r  = wave % tilesPerBatch;
    const int mt = r / ntn;
    const int nt = r % ntn;

    const int lid = lane & 15;              // position within 16-wide half
    const int kh  = lane >> 4;              // which half of the wave

    // A: row (mt*16+lid) of W, this half covers K offsets {kh*2, kh*2+1}
    const float* wrow  = W + (size_t)(mt * 16 + lid) * C256 + kh * 2;
    // B: column (nt*16+lid) of X[b], rows kh*2, kh*2+1 within each K-step
    const float* xbase = X + (size_t)b * C256 * HW + (size_t)(nt * 16 + lid);

    v8f acc = {};
#pragma unroll 4
    for (int kk = 0; kk < C256; kk += 4) {
        v2f a;
        a.x = wrow[kk];
        a.y = wrow[kk + 1];
        const int krow = kk + kh * 2;
        v2f bv;
        bv.x = xbase[(size_t)krow * HW];
        bv.y = xbase[(size_t)(krow + 1) * HW];
        // 8 args: (neg_a, A, neg_b, B, c_mod, C, reuse_a, reuse_b)
        acc = __builtin_amdgcn_wmma_f32_16x16x4_f32(
            false, a, false, bv, (short)0, acc, false, false);
    }

    // D layout: VGPR v -> row (v + kh*8), col lid within the tile
    const int col = nt * 16 + lid;
    float* ybase = Y + (size_t)b * C256 * HW + col;
#pragma unroll
    for (int v = 0; v < 8; ++v) {
        const int row = mt * 16 + v + kh * 8;
        ybase[(size_t)row * HW] = acc[v] + bias[row];
    }
}

// ---------------------------------------------------------------------------
// Kernel 2: 2x2 max-pool, stride 2:  [2,256,56,56] -> [2,256,28,28]
// ---------------------------------------------------------------------------
__global__ void __launch_bounds__(256)
pool2x2_kernel(const float* __restrict__ srcf, float* __restrict__ srcp) {
    const int e = blockIdx.x * 256 + threadIdx.x;
    if (e >= BATCH * C256 * HW_POOL) return;
    const int j  = e % 28;
    const int t  = e / 28;
    const int i  = t % 28;
    const int bc = t / 28;
    const float* p = srcf + ((size_t)bc * 56 + 2 * i) * 56 + 2 * j;
    srcp[e] = fmaxf(fmaxf(p[0], p[1]), fmaxf(p[56], p[57]));
}

// ---------------------------------------------------------------------------
// Kernel 3: per-pixel channel energies
//   s2[b,n] = sum_c src_p[b,c,n]^2 ;  t2[b,n] = sum_k tgt_f[b,k,n]^2
// ---------------------------------------------------------------------------
__global__ void __launch_bounds__(256)
energy_kernel(const float* __restrict__ srcp, const float* __restrict__ tgtf,
              float* __restrict__ s2, float* __restrict__ t2) {
    const int e = blockIdx.x * 256 + threadIdx.x;
    if (e >= BATCH * HW_POOL) return;
    const int b = e / HW_POOL;
    const int n = e % HW_POOL;
    const float* ps = srcp + (size_t)b * C256 * HW_POOL + n;
    const float* pt = tgtf + (size_t)b * C256 * HW_POOL + n;
    float ss = 0.f, tt = 0.f;
#pragma unroll 4
    for (int c = 0; c < C256; ++c) {
        const float v = ps[(size_t)c * HW_POOL];
        const float w = pt[(size_t)c * HW_POOL];
        ss += v * v;
        tt += w * w;
    }
    s2[e] = ss;
    t2[e] = tt;
}

// ---------------------------------------------------------------------------
// Kernel 4: global Frobenius norm of the (never-materialized) correlation
// volume:  norm = sqrt( sum_{b,n} s2[b,n] * t2[b,n] ).  Single block.
// ---------------------------------------------------------------------------
__global__ void __launch_bounds__(256)
norm_kernel(const float* __restrict__ s2, const float* __restrict__ t2,
            float* __restrict__ stats, int n) {
    __shared__ float red[256];
    float sum = 0.f;
    for (int i = threadIdx.x; i < n; i += 256) sum += s2[i] * t2[i];
    red[threadIdx.x] = sum;
    __syncthreads();
    for (int s = 128; s > 0; s >>= 1) {
        if ((int)threadIdx.x < s) red[threadIdx.x] += red[threadIdx.x + s];
        __syncthreads();
    }
    if (threadIdx.x == 0) {
        const float norm = sqrtf(red[0]);
        stats[0] = norm;
        stats[1] = 1.0f / norm;
    }
}

// ---------------------------------------------------------------------------
// Kernel 5: out[b,k,n] = target_x[b,k,n] * (1 + s2[b,n]*tgt_f[b,k,n]/norm)
// ---------------------------------------------------------------------------
__global__ void __launch_bounds__(256)
final_kernel(const float* __restrict__ tgt_x, const float* __restrict__ tgtf,
             const float* __restrict__ s2, const float* __restrict__ stats,
             float* __restrict__ out) {
    const int e = blockIdx.x * 256 + threadIdx.x;
    if (e >= BATCH * C256 * HW_POOL) return;
    const int n  = e % HW_POOL;
    const int b  = e / (C256 * HW_POOL);
    const float inv = stats[1];
    const float cf  = s2[b * HW_POOL + n] * tgtf[e] * inv;
    out[e] = tgt_x[e] * (1.0f + cf);
}

extern "C" void kernel_launch(void* const* d_in, const int* in_sizes, int n_in,
                              void* d_out, int out_size, void* d_ws, size_t ws_size,
                              hipStream_t stream) {
    const float* src_x = (const float*)d_in[0];   // [2,256,56,56]
    const float* tgt_x = (const float*)d_in[1];   // [2,256,28,28]
    const float* src_w = (const float*)d_in[2];   // [256,256] row-major [o,c]
    const float* src_b = (const float*)d_in[3];   // [256]
    const float* tgt_w = (const float*)d_in[4];   // [256,256]
    const float* tgt_b = (const float*)d_in[5];   // [256]
    float* out = (float*)d_out;                   // [2,256,28,28]

    // Workspace layout (floats)
    float* ws    = (float*)d_ws;
    float* src_f = ws;                                        // 2*256*3136
    float* tgt_f = src_f + (size_t)BATCH * C256 * HW_SRC;     // 2*256*784
    float* src_p = tgt_f + (size_t)BATCH * C256 * HW_POOL;    // 2*256*784
    float* s2    = src_p + (size_t)BATCH * C256 * HW_POOL;    // 2*784
    float* t2    = s2 + BATCH * HW_POOL;                      // 2*784
    float* stats = t2 + BATCH * HW_POOL;                      // 2

    // GEMM 1: src_f = src_w @ src_x + src_b   (HW=3136, 196 N-tiles)
    {
        const int ntn = HW_SRC / 16;                       // 196
        const int waves = BATCH * (C256 / 16) * ntn;       // 6272
        gemm_wmma_f32<<<waves * 32 / 256, 256, 0, stream>>>(
            src_x, src_w, src_b, src_f, HW_SRC, ntn);
    }
    // GEMM 2: tgt_f = tgt_w @ target_x + tgt_b   (HW=784, 49 N-tiles)
    {
        const int ntn = HW_POOL / 16;                      // 49
        const int waves = BATCH * (C256 / 16) * ntn;       // 1568
        gemm_wmma_f32<<<waves * 32 / 256, 256, 0, stream>>>(
            tgt_x, tgt_w, tgt_b, tgt_f, HW_POOL, ntn);
    }
    // Max-pool src_f -> src_p
    {
        const int n = BATCH * C256 * HW_POOL;              // 401408
        pool2x2_kernel<<<(n + 255) / 256, 256, 0, stream>>>(src_f, src_p);
    }
    // Channel energies
    {
        const int n = BATCH * HW_POOL;                     // 1568
        energy_kernel<<<(n + 255) / 256, 256, 0, stream>>>(src_p, tgt_f, s2, t2);
    }
    // Global norm
    norm_kernel<<<1, 256, 0, stream>>>(s2, t2, stats, BATCH * HW_POOL);
    // Final scale
    {
        const int n = BATCH * C256 * HW_POOL;
        final_kernel<<<(n + 255) / 256, 256, 0, stream>>>(tgt_x, tgt_f, s2, stats, out);
    }
}